// HybridQCNNQLSTMQuantum_65481071402167
// MI455X (gfx1250) — compile-verified
//
#include <hip/hip_runtime.h>
#include <math.h>

typedef __attribute__((ext_vector_type(2))) float v2f;
typedef __attribute__((ext_vector_type(8))) float v8f;

#define NQ 8
#define DIM 256
#define NBLK 14
#define NLQ 4
#define BATCH 16
#define SEQ 128
#define WAVES_PER_BLOCK 8

struct cplx { float re, im; };
__device__ __forceinline__ cplx cmul(cplx a, cplx b) {
    return { a.re * b.re - a.im * b.im, a.re * b.im + a.im * b.re };
}
__device__ __forceinline__ cplx cadd(cplx a, cplx b) { return { a.re + b.re, a.im + b.im }; }

// ---------------------------------------------------------------------------
// Kernel 0: compose the 14 fused 2-qubit block unitaries (4x4 complex each)
// from the 42 shared QCNN weights, expand to 8x8 real, store to workspace.
// Basis index p = 2*bit(q1) + bit(q2)  (q1 = "msb" of pair subspace).
// ---------------------------------------------------------------------------
__device__ void lmul_msb(cplx (&M)[4][4], const cplx U[2][2]) {
    for (int v = 0; v < 2; ++v)
        for (int j = 0; j < 4; ++j) {
            cplx r0 = M[v][j], r1 = M[2 + v][j];
            M[v][j]     = cadd(cmul(U[0][0], r0), cmul(U[0][1], r1));
            M[2 + v][j] = cadd(cmul(U[1][0], r0), cmul(U[1][1], r1));
        }
}
__device__ void lmul_lsb(cplx (&M)[4][4], const cplx U[2][2]) {
    for (int u = 0; u < 2; ++u)
        for (int j = 0; j < 4; ++j) {
            cplx r0 = M[2 * u][j], r1 = M[2 * u + 1][j];
            M[2 * u][j]     = cadd(cmul(U[0][0], r0), cmul(U[0][1], r1));
            M[2 * u + 1][j] = cadd(cmul(U[1][0], r0), cmul(U[1][1], r1));
        }
}
__device__ void swap_rows(cplx (&M)[4][4], int r0, int r1) {
    for (int j = 0; j < 4; ++j) { cplx t = M[r0][j]; M[r0][j] = M[r1][j]; M[r1][j] = t; }
}
__device__ void mk_rz(cplx U[2][2], float t) {
    float h = 0.5f * t;
    U[0][0] = { __cosf(h), -__sinf(h) }; U[0][1] = { 0.f, 0.f };
    U[1][0] = { 0.f, 0.f };              U[1][1] = { __cosf(h), __sinf(h) };
}
__device__ void mk_ry(cplx U[2][2], float t) {
    float c = __cosf(0.5f * t), s = __sinf(0.5f * t);
    U[0][0] = { c, 0.f }; U[0][1] = { -s, 0.f };
    U[1][0] = { s, 0.f }; U[1][1] = { c, 0.f };
}

__global__ void fuse_blocks_kernel(const float* __restrict__ w, float* __restrict__ fused) {
    int blk = threadIdx.x;
    if (blk >= NBLK) return;
    const int poff_tab[NBLK] = {0,3,6,9,12,15,18,21,24,27,30,33,36,39};
    const int is_conv[NBLK]  = {1,1,1,1,0,0,0,0,1,1,0,0,1,0};
    float p0 = w[poff_tab[blk] + 0];
    float p1 = w[poff_tab[blk] + 1];
    float p2 = w[poff_tab[blk] + 2];
    const float HPI = 1.57079632679489662f;

    cplx M[4][4];
    for (int i = 0; i < 4; ++i)
        for (int j = 0; j < 4; ++j) M[i][j] = { (i == j) ? 1.f : 0.f, 0.f };

    cplx U[2][2];
    // shared prefix: RZ(-pi/2) on q2 ; CX(q2->q1) ; RZ(p0) q1 ; RY(p1) q2 ; CX(q1->q2) ; RY(p2) q2
    mk_rz(U, -HPI);  lmul_lsb(M, U);
    swap_rows(M, 1, 3);            // CX control=lsb(q2), target=msb(q1)
    mk_rz(U, p0);    lmul_msb(M, U);
    mk_ry(U, p1);    lmul_lsb(M, U);
    swap_rows(M, 2, 3);            // CX control=msb(q1), target=lsb(q2)
    mk_ry(U, p2);    lmul_lsb(M, U);
    if (is_conv[blk]) {            // conv tail: CX(q2->q1) ; RZ(pi/2) q1
        swap_rows(M, 1, 3);
        mk_rz(U, HPI); lmul_msb(M, U);
    }

    // expand to 8x8 real (interleaved re/im), row-major, 64 floats per block
    float* G = fused + blk * 64;
    for (int p = 0; p < 4; ++p)
        for (int q = 0; q < 4; ++q) {
            cplx m = M[p][q];
            G[(2 * p) * 8 + 2 * q]         =  m.re;
            G[(2 * p) * 8 + 2 * q + 1]     = -m.im;
            G[(2 * p + 1) * 8 + 2 * q]     =  m.im;
            G[(2 * p + 1) * 8 + 2 * q + 1] =  m.re;
        }
}

// ---------------------------------------------------------------------------
// Kernel 1: QCNN. One wave32 per sample; statevector (256 x float2) in LDS.
// Fused 2-qubit blocks applied with V_WMMA_F32_16X16X4_F32 (K=8 split in two
// accumulating K=4 WMMAs). 112 WMMA per sample.
// ---------------------------------------------------------------------------
__constant__ int c_qa[NBLK] = {0,2,4,6,0,2,4,6,0,2,0,2,0,0};

__device__ __forceinline__ int expand2(int o, int hi, int lo) {
    int t = ((o >> lo) << (lo + 1)) | (o & ((1 << lo) - 1));
    t = ((t >> hi) << (hi + 1)) | (t & ((1 << hi) - 1));
    return t;
}

__global__ __launch_bounds__(256) void qcnn_kernel(const float* __restrict__ x,
                                                   const float* __restrict__ fused,
                                                   float* __restrict__ out) {
    __shared__ float2 st_sh[WAVES_PER_BLOCK][DIM];
    const int lane = threadIdx.x & 31;
    const int wv   = threadIdx.x >> 5;
    const int samp = blockIdx.x * WAVES_PER_BLOCK + wv;
    float2* st = st_sh[wv];

    float xv[NQ];
    const float* xs = x + samp * NQ;
    #pragma unroll
    for (int q = 0; q < NQ; ++q) xv[q] = xs[q];

    const float inv16 = 0.0625f;
    const float invs2 = 0.70710678118654752f;

    // rep 1 folded: H^8|0> = uniform 1/16, then diag phase exp(i*2*sum x_q b_q)
    #pragma unroll
    for (int r = 0; r < 8; ++r) {
        int idx = lane * 8 + r;
        float phi = 0.f;
        #pragma unroll
        for (int q = 0; q < NQ; ++q)
            if ((idx >> (7 - q)) & 1) phi += 2.f * xv[q];
        st[idx] = make_float2(inv16 * __cosf(phi), inv16 * __sinf(phi));
    }

    // rep 2: H on every qubit (butterflies over each bit position)
    for (int bp = 0; bp < 8; ++bp) {
        int lowmask = (1 << bp) - 1;
        #pragma unroll
        for (int r = 0; r < 4; ++r) {
            int j  = lane + 32 * r;                       // pair 0..127
            int i0 = ((j & ~lowmask) << 1) | (j & lowmask);
            int i1 = i0 | (1 << bp);
            float2 a = st[i0], b = st[i1];
            st[i0] = make_float2((a.x + b.x) * invs2, (a.y + b.y) * invs2);
            st[i1] = make_float2((a.x - b.x) * invs2, (a.y - b.y) * invs2);
        }
    }
    // rep 2 diag phase layer
    #pragma unroll
    for (int r = 0; r < 8; ++r) {
        int idx = lane * 8 + r;
        float phi = 0.f;
        #pragma unroll
        for (int q = 0; q < NQ; ++q)
            if ((idx >> (7 - q)) & 1) phi += 2.f * xv[q];
        float2 a = st[idx];
        float cph = __cosf(phi), sph = __sinf(phi);
        st[idx] = make_float2(a.x * cph - a.y * sph, a.x * sph + a.y * cph);
    }

    // 14 fused two-qubit blocks via WMMA f32 16x16x4
    const int m    = lane & 15;
    const int half = lane >> 4;    // 0: K 0..1, 1: K 2..3 (per ISA A-frag layout)
    const int n    = lane & 15;    // B/D column

    for (int blk = 0; blk < NBLK; ++blk) {
        int a = c_qa[blk];
        int hi = 7 - a;            // flat-bit of qubit a (pair msb)
        int lo = 6 - a;            // flat-bit of qubit a+1 (pair lsb)
        const float* G = fused + blk * 64;

        // A fragments: rows 0..7 = 8x8 real gate, rows 8..15 = 0
        float rowok = (m < 8) ? 1.f : 0.f;
        int gm = (m & 7) * 8;
        int k0 = half * 2;
        v2f A0, A1;
        A0.x = rowok * G[gm + k0];     A0.y = rowok * G[gm + k0 + 1];
        A1.x = rowok * G[gm + 4 + k0]; A1.y = rowok * G[gm + 4 + k0 + 1];

        for (int t = 0; t < 4; ++t) {                 // 4 tiles x 16 groups = 64
            int base = expand2(16 * t + n, hi, lo);
            // chunk 0 holds complex p = half, chunk 1 holds p = 2 + half
            int i0 = base | (half << lo);                 // p = half  -> bits (0, half)
            int i1 = base | (1 << hi) | (half << lo);     // p = 2+half-> bits (1, half)
            float2 amp0 = st[i0];
            float2 amp1 = st[i1];
            v2f B0; B0.x = amp0.x; B0.y = amp0.y;
            v2f B1; B1.x = amp1.x; B1.y = amp1.y;

            v8f acc = {0.f, 0.f, 0.f, 0.f, 0.f, 0.f, 0.f, 0.f};
            acc = __builtin_amdgcn_wmma_f32_16x16x4_f32(false, A0, false, B0,
                                                        (short)0, acc, false, false);
            acc = __builtin_amdgcn_wmma_f32_16x16x4_f32(false, A1, false, B1,
                                                        (short)0, acc, false, false);

            if (lane < 16) {   // lanes 0..15 hold D rows 0..7 (the useful data)
                int wb = expand2(16 * t + lane, hi, lo);
                #pragma unroll
                for (int p = 0; p < 4; ++p) {
                    int widx = wb | ((p >> 1) << hi) | ((p & 1) << lo);
                    st[widx] = make_float2(acc[2 * p], acc[2 * p + 1]);
                }
            }
        }
    }

    // <Z_7>: sign by flat LSB; wave32 reduction
    float z = 0.f;
    #pragma unroll
    for (int r = 0; r < 8; ++r) {
        int idx = lane * 8 + r;
        float2 aamp = st[idx];
        float pr = aamp.x * aamp.x + aamp.y * aamp.y;
        z += (idx & 1) ? -pr : pr;
    }
    #pragma unroll
    for (int off = 16; off > 0; off >>= 1) z += __shfl_xor(z, off, 32);
    if (lane == 0) out[samp] = z;
}

// ---------------------------------------------------------------------------
// Kernel 2: QLSTM scan. 64 threads = 4 gates x 16 samples. 4-qubit state per
// thread fully in registers.
// ---------------------------------------------------------------------------
__device__ __forceinline__ void rx_gate(float2 st[16], float t, int bp) {
    float s, c;
    __sincosf(0.5f * t, &s, &c);
    int msk = 1 << bp;
    #pragma unroll
    for (int i = 0; i < 16; ++i) {
        if (i & msk) continue;
        float2 a = st[i], d = st[i | msk];
        st[i]       = make_float2(c * a.x + s * d.y, c * a.y - s * d.x);
        st[i | msk] = make_float2(c * d.x + s * a.y, c * d.y - s * a.x);
    }
}
__device__ __forceinline__ void cx_gate(float2 st[16], int cb, int tb) {
    int cm = 1 << cb, tm = 1 << tb;
    #pragma unroll
    for (int i = 0; i < 16; ++i)
        if ((i & cm) && !(i & tm)) { float2 t = st[i]; st[i] = st[i | tm]; st[i | tm] = t; }
}
__device__ __forceinline__ float sigf(float v) { return 1.f / (1.f + __expf(-v)); }

__global__ __launch_bounds__(64) void qlstm_kernel(
    const float* __restrict__ qcnn_out,
    const float* thf, const float* thi, const float* thg, const float* tho,
    const float* Wf_, const float* bf_, const float* Wi_, const float* bi_,
    const float* Wg_, const float* bg_, const float* Wo_, const float* bo_,
    const float* Wh_, const float* bh_, float* __restrict__ logits) {
    __shared__ float h_sh[BATCH][NLQ];
    __shared__ float c_sh[BATCH][NLQ];
    __shared__ float e_sh[4][BATCH][NLQ];

    const int tid = threadIdx.x;
    const int g = tid >> 4;      // gate: 0=f 1=i 2=g 3=o
    const int b = tid & 15;      // sample

    const float* Wtab[4] = { Wf_, Wi_, Wg_, Wo_ };
    const float* btab[4] = { bf_, bi_, bg_, bo_ };
    const float* ttab[4] = { thf, thi, thg, tho };

    float W[4][5], bb[4], th[4];
    #pragma unroll
    for (int j = 0; j < 4; ++j) {
        bb[j] = btab[g][j];
        th[j] = ttab[g][j];
        #pragma unroll
        for (int k = 0; k < 5; ++k) W[j][k] = Wtab[g][j * 5 + k];
    }

    if (tid < BATCH)
        for (int j = 0; j < NLQ; ++j) { h_sh[tid][j] = 0.f; c_sh[tid][j] = 0.f; }
    __syncthreads();

    for (int t = 0; t < SEQ; ++t) {
        float xt = qcnn_out[b * SEQ + t];
        float v[4];
        #pragma unroll
        for (int j = 0; j < 4; ++j) {
            float zz = bb[j] + W[j][0] * xt;
            #pragma unroll
            for (int k = 0; k < 4; ++k) zz += W[j][k + 1] * h_sh[b][k];
            v[j] = zz;
        }

        // 4-qubit circuit: RX(v_q) q, RX(th_q) q, CX ring
        float2 st[16];
        st[0] = make_float2(1.f, 0.f);
        #pragma unroll
        for (int i = 1; i < 16; ++i) st[i] = make_float2(0.f, 0.f);
        #pragma unroll
        for (int q = 0; q < 4; ++q) rx_gate(st, v[q], 3 - q);
        #pragma unroll
        for (int q = 0; q < 4; ++q) rx_gate(st, th[q], 3 - q);
        cx_gate(st, 3, 2); cx_gate(st, 2, 1); cx_gate(st, 1, 0); cx_gate(st, 0, 3);

        #pragma unroll
        for (int q = 0; q < 4; ++q) {
            int bp = 3 - q;
            float e = 0.f;
            #pragma unroll
            for (int i = 0; i < 16; ++i) {
                float pr = st[i].x * st[i].x + st[i].y * st[i].y;
                e += (i & (1 << bp)) ? -pr : pr;
            }
            e_sh[g][b][q] = e;
        }
        __syncthreads();

        if (tid < BATCH) {
            #pragma unroll
            for (int j = 0; j < NLQ; ++j) {
                float f  = sigf(e_sh[0][tid][j]);
                float ii = sigf(e_sh[1][tid][j]);
                float gg = tanhf(e_sh[2][tid][j]);
                float o  = sigf(e_sh[3][tid][j]);
                float cn = f * c_sh[tid][j] + ii * gg;
                c_sh[tid][j] = cn;
                h_sh[tid][j] = o * tanhf(cn);
            }
        }
        __syncthreads();
    }

    if (tid < BATCH) {
        float lg = bh_[0];
        #pragma unroll
        for (int j = 0; j < NLQ; ++j) lg += h_sh[tid][j] * Wh_[j];
        logits[tid] = lg;
    }
}

// ---------------------------------------------------------------------------
extern "C" void kernel_launch(void* const* d_in, const int* in_sizes, int n_in,
                              void* d_out, int out_size, void* d_ws, size_t ws_size,
                              hipStream_t stream) {
    const float* x      = (const float*)d_in[0];   // (16,128,8)
    const float* qcnn_w = (const float*)d_in[1];   // (42,)
    const float* thf    = (const float*)d_in[2];
    const float* thi    = (const float*)d_in[3];
    const float* thg    = (const float*)d_in[4];
    const float* tho    = (const float*)d_in[5];
    const float* Wf     = (const float*)d_in[6];
    const float* bf     = (const float*)d_in[7];
    const float* Wi     = (const float*)d_in[8];
    const float* bi     = (const float*)d_in[9];
    const float* Wg     = (const float*)d_in[10];
    const float* bg     = (const float*)d_in[11];
    const float* Wo     = (const float*)d_in[12];
    const float* bo     = (const float*)d_in[13];
    const float* Wh     = (const float*)d_in[14];
    const float* bh     = (const float*)d_in[15];

    float* out   = (float*)d_out;        // [0..2047] qcnn_out, [2048..2063] logits
    float* fused = (float*)d_ws;         // 14 * 64 floats

    fuse_blocks_kernel<<<1, 32, 0, stream>>>(qcnn_w, fused);
    qcnn_kernel<<<2048 / WAVES_PER_BLOCK, 32 * WAVES_PER_BLOCK, 0, stream>>>(x, fused, out);
    qlstm_kernel<<<1, 64, 0, stream>>>(out, thf, thi, thg, tho,
                                       Wf, bf, Wi, bi, Wg, bg, Wo, bo,
                                       Wh, bh, out + BATCH * SEQ);
}